// GraphConvLayer_38422777430231
// MI455X (gfx1250) — compile-verified
//
#include <hip/hip_runtime.h>

// Problem constants (from reference): B=4, T=4, N=256, D=128
#define BT 16
#define NN 256
#define DD 128
#define DP (DD + 4)   // padded row (LDS bank-conflict avoidance)
#define NP (NN + 4)   // padded score row
#define EPSV 1e-5f
#define NEGV -1000000000.0f

typedef __attribute__((ext_vector_type(2))) float v2f;
typedef __attribute__((ext_vector_type(4))) float v4f;
typedef __attribute__((ext_vector_type(8))) float v8f;

// D = A(16x4 f32) * B(4x16 f32) + C(16x16 f32); wave32, full-precision f32 path.
// A frag: lane L holds A[L&15][2*(L>>4)+{0,1}]  (v2f)
// B frag: lane L holds B[2*(L>>4)+{0,1}][L&15]  (v2f)
// C/D:    VGPR r, lane L holds D[r + 8*(L>>4)][L&15]
__device__ __forceinline__ v8f wmma4(v2f a, v2f b, v8f c) {
  return __builtin_amdgcn_wmma_f32_16x16x4_f32(
      /*neg_a=*/false, a, /*neg_b=*/false, b,
      /*c_mod=*/(short)0, c, /*reuse_a=*/false, /*reuse_b=*/false);
}

// ---------------------------------------------------------------------------
// K1: H = X@W + bias ; Aatt = H@W1a ; Batt = H@W1b   (all [BT*N, D] f32)
// grid = 64 blocks x 128 threads (4 waves); each wave owns a 16-row block.
// Fully-unrolled K loops (A-frags stay in VGPRs); dual accumulators break
// the WMMA RAW chain; hTile padded to kill 16-way LDS bank conflicts.
// ---------------------------------------------------------------------------
__global__ __launch_bounds__(128) void k_gemm3(
    const float* __restrict__ X, const float* __restrict__ W,
    const float* __restrict__ bias, const float* __restrict__ W1,
    float* __restrict__ H, float* __restrict__ Aatt, float* __restrict__ Batt) {
  __shared__ __align__(16) float hTile[4][16][DP];  // 33 KB
  const int wave = threadIdx.x >> 5;
  const int lane = threadIdx.x & 31;
  const int m  = lane & 15;
  const int hi = lane >> 4;
  const int rowBase = (blockIdx.x * 4 + wave) * 16;  // in [0, BT*NN)

  // A fragments of the X row-block for the full K=128 (32 k-steps) - in VGPRs
  v2f ax[32];
#pragma unroll
  for (int k = 0; k < 32; ++k)
    ax[k] = *(const v2f*)&X[(rowBase + m) * DD + 4 * k + 2 * hi];

  // ---- H = X*W + bias, 8 N-tiles ----
#pragma unroll 1
  for (int nt = 0; nt < 8; ++nt) {
    v8f acc0 = {}, acc1 = {};
    const float* wp = W + nt * 16 + m;  // column index == lane&15
#pragma unroll
    for (int k = 0; k < 32; k += 2) {
      v2f b0, b1;
      b0.x = wp[(4 * k + 2 * hi) * DD];
      b0.y = wp[(4 * k + 2 * hi + 1) * DD];
      b1.x = wp[(4 * k + 4 + 2 * hi) * DD];
      b1.y = wp[(4 * k + 5 + 2 * hi) * DD];
      acc0 = wmma4(ax[k], b0, acc0);
      acc1 = wmma4(ax[k + 1], b1, acc1);
    }
    const float bv = bias[nt * 16 + m];
#pragma unroll
    for (int r = 0; r < 8; ++r) {
      const float v = acc0[r] + acc1[r] + bv;
      H[(rowBase + r + 8 * hi) * DD + nt * 16 + m] = v;
      hTile[wave][r + 8 * hi][nt * 16 + m] = v;
    }
  }
  __syncthreads();

  // Re-read own h tile in A-fragment layout (registers, full unroll)
  v2f ah[32];
#pragma unroll
  for (int k = 0; k < 32; ++k)
    ah[k] = *(const v2f*)&hTile[wave][m][4 * k + 2 * hi];

  // ---- Aatt = H*W1a ; Batt = H*W1b ----
#pragma unroll 1
  for (int g = 0; g < 2; ++g) {
    const float* Wg = W1 + g * DD * DD;     // att_w1[:D] then att_w1[D:]
    float* Og = g ? Batt : Aatt;
#pragma unroll 1
    for (int nt = 0; nt < 8; ++nt) {
      v8f acc0 = {}, acc1 = {};
      const float* wp = Wg + nt * 16 + m;
#pragma unroll
      for (int k = 0; k < 32; k += 2) {
        v2f b0, b1;
        b0.x = wp[(4 * k + 2 * hi) * DD];
        b0.y = wp[(4 * k + 2 * hi + 1) * DD];
        b1.x = wp[(4 * k + 4 + 2 * hi) * DD];
        b1.y = wp[(4 * k + 5 + 2 * hi) * DD];
        acc0 = wmma4(ah[k], b0, acc0);
        acc1 = wmma4(ah[k + 1], b1, acc1);
      }
#pragma unroll
      for (int r = 0; r < 8; ++r)
        Og[(rowBase + r + 8 * hi) * DD + nt * 16 + m] = acc0[r] + acc1[r];
    }
  }
}

// ---------------------------------------------------------------------------
// K2: pairwise leaky-relu scores + mask + softmax + agg = attn@H + LN + ReLU
// grid = BT*16 blocks (bt, 16-row i-block) x 256 threads (8 waves).
// Batt staged through LDS in 64-row tiles (16x less L2 traffic); sAgg aliases
// the Batt tile storage (barrier-separated) to stay under 64 KB static LDS.
// ---------------------------------------------------------------------------
__global__ __launch_bounds__(256) void k_attn(
    const float* __restrict__ X, const float* __restrict__ adj,
    const float* __restrict__ att_b1, const float* __restrict__ att_w2,
    const float* __restrict__ att_b2, const float* __restrict__ lng,
    const float* __restrict__ lnb,
    const float* __restrict__ H, const float* __restrict__ Aatt,
    const float* __restrict__ Batt, float* __restrict__ Out) {
  __shared__ __align__(16) float sA[16][DD];     // a-rows of this i-block (8 KB)
  __shared__ __align__(16) float sScore[16][NP]; // scores -> attn (16.25 KB, padded)
  __shared__ __align__(16) float sB[64][DP];     // Batt tile (33 KB, padded)
  __shared__ __align__(16) float sC[DD];         // att_b1
  __shared__ __align__(16) float sW2[DD];        // att_w2[:,0]
  float (*sAgg)[DP] = sB;                        // agg reuses sB storage

  const int bt = blockIdx.x >> 4;
  const int ib = blockIdx.x & 15;
  const int tid = threadIdx.x;
  const int wave = tid >> 5;
  const int lane = tid & 31;

  for (int idx = tid; idx < 16 * 32; idx += 256) {  // 16x128 as v4f
    const int r = idx >> 5, dq = (idx & 31) * 4;
    *(v4f*)&sA[r][dq] = *(const v4f*)&Aatt[(bt * NN + ib * 16 + r) * DD + dq];
  }
  if (tid < DD) { sC[tid] = att_b1[tid]; sW2[tid] = att_w2[tid]; }
  __syncthreads();

  const float b2 = att_b2[0];

  // ---- scores[i][j] = sum_d lrelu(a[i,d]+b[j,d]+c[d]) * w2[d]  (VALU) ----
#pragma unroll 1
  for (int jt = 0; jt < 4; ++jt) {                 // 4 tiles of 64 j-rows
    for (int idx = tid; idx < 64 * 32; idx += 256) {  // stage 64x128 as v4f
      const int r = idx >> 5, dq = (idx & 31) * 4;
      *(v4f*)&sB[r][dq] = *(const v4f*)&Batt[(bt * NN + jt * 64 + r) * DD + dq];
    }
    __syncthreads();
#pragma unroll 1
    for (int ii = 0; ii < 2; ++ii) {
      const int i = wave * 2 + ii;
      const int gi = ib * 16 + i;
#pragma unroll 1
      for (int jh = 0; jh < 2; ++jh) {
        const int jl = jh * 32 + lane;
        const int j = jt * 64 + jl;
        const float* bp = sB[jl];
        float s = 0.f;
#pragma unroll 8
        for (int d = 0; d < DD; d += 4) {
          v4f av = *(const v4f*)&sA[i][d];
          v4f bv = *(const v4f*)&bp[d];
          v4f cv = *(const v4f*)&sC[d];
          v4f wv = *(const v4f*)&sW2[d];
          v4f z = av + bv + cv;
          z.x = fmaxf(z.x, 0.2f * z.x);  // leaky_relu(z,0.2) = max(z, 0.2z)
          z.y = fmaxf(z.y, 0.2f * z.y);
          z.z = fmaxf(z.z, 0.2f * z.z);
          z.w = fmaxf(z.w, 0.2f * z.w);
          s += z.x * wv.x + z.y * wv.y + z.z * wv.z + z.w * wv.w;
        }
        s += b2;
        if (adj[gi * NN + j] == 0.f) s = NEGV;  // mask
        sScore[i][j] = s;
      }
    }
    __syncthreads();  // protect sB before next tile overwrites it
  }

  // ---- softmax over j (wave32 butterflies) ----
#pragma unroll 1
  for (int ii = 0; ii < 2; ++ii) {
    const int i = wave * 2 + ii;
    float mx = -3.4e38f;
#pragma unroll
    for (int jj = 0; jj < 8; ++jj) mx = fmaxf(mx, sScore[i][jj * 32 + lane]);
    for (int off = 16; off; off >>= 1) mx = fmaxf(mx, __shfl_xor(mx, off, 32));
    float e[8], sum = 0.f;
#pragma unroll
    for (int jj = 0; jj < 8; ++jj) {
      e[jj] = __expf(sScore[i][jj * 32 + lane] - mx);
      sum += e[jj];
    }
    for (int off = 16; off; off >>= 1) sum += __shfl_xor(sum, off, 32);
    const float inv = 1.f / sum;
#pragma unroll
    for (int jj = 0; jj < 8; ++jj) sScore[i][jj * 32 + lane] = e[jj] * inv;
  }
  __syncthreads();

  // ---- agg = attn(16x256) @ H(256x128): wave w owns N-tile w; K=256 ----
  const int m = lane & 15;
  const int hi = lane >> 4;
  const int nt = wave;
  v8f acc0 = {}, acc1 = {};
#pragma unroll 8
  for (int k = 0; k < 64; k += 2) {
    v2f a0 = *(const v2f*)&sScore[m][4 * k + 2 * hi];
    v2f a1 = *(const v2f*)&sScore[m][4 * k + 4 + 2 * hi];
    const float* hp = &H[(bt * NN + 4 * k + 2 * hi) * DD + nt * 16 + m];
    v2f b0, b1;
    b0.x = hp[0];
    b0.y = hp[DD];
    b1.x = hp[4 * DD];
    b1.y = hp[5 * DD];
    acc0 = wmma4(a0, b0, acc0);
    acc1 = wmma4(a1, b1, acc1);
  }
#pragma unroll
  for (int r = 0; r < 8; ++r)
    sAgg[r + 8 * hi][nt * 16 + m] = acc0[r] + acc1[r];
  __syncthreads();

  // ---- residual + LayerNorm + ReLU (one row per wave pass) ----
#pragma unroll 1
  for (int ii = 0; ii < 2; ++ii) {
    const int i = wave * 2 + ii;
    const int gi = ib * 16 + i;
    const int d = lane * 4;
    v4f y = *(const v4f*)&sAgg[i][d];
    v4f xr = *(const v4f*)&X[(bt * NN + gi) * DD + d];
    y += xr;
    float p = y.x + y.y + y.z + y.w;
    for (int off = 16; off; off >>= 1) p += __shfl_xor(p, off, 32);
    const float mu = p * (1.f / 128.f);
    v4f yc = y - mu;
    float q = yc.x * yc.x + yc.y * yc.y + yc.z * yc.z + yc.w * yc.w;
    for (int off = 16; off; off >>= 1) q += __shfl_xor(q, off, 32);
    const float rs = rsqrtf(q * (1.f / 128.f) + EPSV);
    v4f g = *(const v4f*)&lng[d];
    v4f bb = *(const v4f*)&lnb[d];
    v4f o = yc * rs * g + bb;
    o.x = fmaxf(o.x, 0.f);
    o.y = fmaxf(o.y, 0.f);
    o.z = fmaxf(o.z, 0.f);
    o.w = fmaxf(o.w, 0.f);
    *(v4f*)&Out[(bt * NN + gi) * DD + d] = o;
  }
}

// ---------------------------------------------------------------------------
extern "C" void kernel_launch(void* const* d_in, const int* in_sizes, int n_in,
                              void* d_out, int out_size, void* d_ws, size_t ws_size,
                              hipStream_t stream) {
  const float* x       = (const float*)d_in[0];  // (B,T,N,D)
  const float* adj     = (const float*)d_in[1];  // (N,N)
  const float* weight  = (const float*)d_in[2];  // (D,D)
  const float* bias    = (const float*)d_in[3];  // (D)
  const float* att_w1  = (const float*)d_in[4];  // (2D,D)
  const float* att_b1  = (const float*)d_in[5];  // (D)
  const float* att_w2  = (const float*)d_in[6];  // (D,1)
  const float* att_b2  = (const float*)d_in[7];  // (1)
  const float* ln_g    = (const float*)d_in[8];  // (D)
  const float* ln_b    = (const float*)d_in[9];  // (D)
  float* out = (float*)d_out;

  float* H    = (float*)d_ws;            // [BT*N*D]
  float* Aatt = H + BT * NN * DD;        // [BT*N*D]
  float* Batt = Aatt + BT * NN * DD;     // [BT*N*D]  (6 MB total workspace)

  k_gemm3<<<dim3(64), dim3(128), 0, stream>>>(x, weight, bias, att_w1, H, Aatt, Batt);
  k_attn<<<dim3(BT * 16), dim3(256), 0, stream>>>(x, adj, att_b1, att_w2, att_b2,
                                                  ln_g, ln_b, H, Aatt, Batt, out);
  (void)in_sizes; (void)n_in; (void)out_size; (void)ws_size;
}